// InterfaceBoundaryLoss_28303834481397
// MI455X (gfx1250) — compile-verified
//
#include <hip/hip_runtime.h>
#include <hip/hip_bf16.h>

#define NNX 64
#define NNY 64
#define NNZ 64
#define VOL (NNX*NNY*NNZ)

typedef float v2f __attribute__((ext_vector_type(2)));
typedef float v8f __attribute__((ext_vector_type(8)));

#define INV_EPSF 8388608.0f            // 1 / np.finfo(float32).eps = 2^23
#define INV4PI   0.07957747154594767f  // 1/(4*pi), E_IN = 1
#define DXF      (1.0f/63.0f)          // 1/(NNX-1)
#define INVDX    63.0f
#define EOUTF    80.0f
#define CHUNK    256                   // charges staged in LDS per pass

// xor-lane float add via ds_swizzle (group-of-32 bit mode: and=0x1f, or=0, xor in [14:10])
#define XRED(x, imm) (x) += __int_as_float(__builtin_amdgcn_ds_swizzle(__float_as_int(x), (imm)))

__global__ __launch_bounds__(32)
void ibl_greens_kernel(const float* __restrict__ outv,
                       const float* __restrict__ q,
                       const float* __restrict__ xq,
                       const int*   __restrict__ xi,
                       const int*   __restrict__ yi,
                       const int*   __restrict__ zi,
                       const float* __restrict__ normals,
                       float*       __restrict__ partials,
                       int NB, int NQ, int Bn)
{
    __shared__ float4 s_c[CHUNK];        // {x, y, z, q} per charge -> ds_load_b128

    const int lane = threadIdx.x;        // one wave32 per block
    const int half = lane >> 4;          // 0: K=0,1 / M=v      1: K=2,3 / M=v+8
    const int ln   = lane & 15;
    const int n    = blockIdx.x * 16 + ln;        // boundary point index
    const int nc   = (n < NB) ? n : (NB - 1);     // clamped (tail lanes masked later)

    const int ix = xi[nc], iy = yi[nc], iz = zi[nc];
    const float px = ix * DXF, py = iy * DXF, pz = iz * DXF;
    const float pn2 = px*px + py*py + pz*pz;

    // B (4x16 f32): col n = (px,py,pz,1); VGPR0 = rows K0/K2, VGPR1 = rows K1/K3
    v2f Bm;
    Bm.x = half ? pz   : px;
    Bm.y = half ? 1.0f : py;

    float accG = 0.0f;                   // sum q/r
    float accC = 0.0f;                   // sum coef
    float accX = 0.0f, accY = 0.0f, accZ = 0.0f;   // sum coef * x_k

    for (int base = 0; base < NQ; base += CHUNK) {
        int cnt  = NQ - base; if (cnt > CHUNK) cnt = CHUNK;
        int cntp = (cnt + 15) & ~15;                 // pad to 16 (q=0 -> no contribution)
        for (int i = lane; i < cntp; i += 32) {
            bool ok = i < cnt;
            float4 v4;
            v4.x = ok ? xq[3*(base+i) + 0] : 0.0f;
            v4.y = ok ? xq[3*(base+i) + 1] : 0.0f;
            v4.z = ok ? xq[3*(base+i) + 2] : 0.0f;
            v4.w = ok ? q[base+i]          : 0.0f;
            s_c[i] = v4;
        }
        __syncthreads();

        const int nkt = cntp >> 4;
        for (int kt = 0; kt < nkt; ++kt) {
            const int k0 = kt << 4;
            // A (16x4 f32): row k = (-2x, -2y, -2z, |x|^2); this lane supplies row ln
            const float4 xa = s_c[k0 + ln];
            const float x2 = xa.x*xa.x + xa.y*xa.y + xa.z*xa.z;
            v2f Am;
            Am.x = half ? (-2.0f*xa.z) : (-2.0f*xa.x);
            Am.y = half ?  x2          : (-2.0f*xa.y);

            v8f C = {};
            // D(k,n) = -2 x_k . p_n + |x_k|^2   (r^2 tile up to +|p_n|^2)
            v8f D = __builtin_amdgcn_wmma_f32_16x16x4_f32(
                        false, Am, false, Bm, (short)0, C, false, false);

            #pragma unroll
            for (int v = 0; v < 8; ++v) {
                const float4 cq = s_c[k0 + v + (half << 3)];   // one ds_load_b128
                const float  qv = cq.w;
                float r2 = fmaxf(D[v] + pn2, 0.0f);
                const bool z = (r2 == 0.0f);
                float invr  = __builtin_amdgcn_rsqf(r2);       // v_rsq_f32
                invr        = z ? INV_EPSF : invr;             // reference: r==0 -> EPS
                float invr3 = invr * invr * invr;              // 1 / r2^1.5
                invr3       = z ? INV_EPSF : invr3;            // reference: r3==0 -> EPS
                accG += qv * invr;
                const float coef = -qv * invr3;
                accC += coef;
                accX += coef * cq.x;
                accY += coef * cq.y;
                accZ += coef * cq.z;
            }
        }
        __syncthreads();
    }

    // fold the two half-waves: lane n holds charges {v}, lane n+16 holds {v+8}
    XRED(accG, 0x401F); XRED(accC, 0x401F);
    XRED(accX, 0x401F); XRED(accY, 0x401F); XRED(accZ, 0x401F);

    const float g  = accG * INV4PI;
    const float gx = (px * accC - accX) * INV4PI;   // grad_g = (p*Sum(coef) - Sum(coef*x))/4pi
    const float gy = (py * accC - accY) * INV4PI;
    const float gz = (pz * accC - accZ) * INV4PI;
    const float nx = normals[3*nc+0], ny = normals[3*nc+1], nz = normals[3*nc+2];
    const float gc = gx*nx + gy*ny + gz*nz;

    float c1 = g * g;
    float c2 = 0.0f;
    const int idx3 = (ix * NNY + iy) * NNZ + iz;
    for (int b = 0; b < Bn; ++b) {
        const float* ob = outv + (size_t)b * VOL;
        const float cv = ob[idx3];
        const float lx = ob[idx3 - NNY*NNZ], rx = ob[idx3 + NNY*NNZ];
        const float ly = ob[idx3 - NNZ],     ry = ob[idx3 + NNZ];
        const float lz = ob[idx3 - 1],       rz = ob[idx3 + 1];
        const float gxi = (nx > 0.0f) ? (cv-lx)*INVDX : (rx-cv)*INVDX;
        const float gxo = (nx > 0.0f) ? (rx-cv)*INVDX : (cv-lx)*INVDX;
        const float gyi = (ny > 0.0f) ? (cv-ly)*INVDX : (ry-cv)*INVDX;
        const float gyo = (ny > 0.0f) ? (ry-cv)*INVDX : (cv-ly)*INVDX;
        const float gzi = (nz > 0.0f) ? (cv-lz)*INVDX : (rz-cv)*INVDX;
        const float gzo = (nz > 0.0f) ? (rz-cv)*INVDX : (cv-lz)*INVDX;
        const float ndi = gxi*nx + gyi*ny + gzi*nz;
        const float ndo = gxo*nx + gyo*ny + gzo*nz;
        const float t   = (ndi + gc) - EOUTF * ndo;   // E_IN = 1
        c2 += t * t;
    }

    const float act = (half == 0 && n < NB) ? 1.0f : 0.0f;
    float p1 = act * c1;
    float p2 = act * c2;
    XRED(p1, 0x041F); XRED(p2, 0x041F);
    XRED(p1, 0x081F); XRED(p2, 0x081F);
    XRED(p1, 0x101F); XRED(p2, 0x101F);
    XRED(p1, 0x201F); XRED(p2, 0x201F);
    XRED(p1, 0x401F); XRED(p2, 0x401F);
    if (lane == 0) {
        partials[2*blockIdx.x + 0] = p1;
        partials[2*blockIdx.x + 1] = p2;
    }
}

__global__ __launch_bounds__(256)
void ibl_reduce_kernel(const float* __restrict__ partials, int nblocks,
                       int NB, int Bn, float* __restrict__ outp)
{
    __shared__ float s1[256];
    __shared__ float s2[256];
    float a = 0.0f, b = 0.0f;
    for (int i = threadIdx.x; i < nblocks; i += 256) {
        a += partials[2*i + 0];
        b += partials[2*i + 1];
    }
    s1[threadIdx.x] = a;
    s2[threadIdx.x] = b;
    __syncthreads();
    for (int off = 128; off > 0; off >>= 1) {
        if (threadIdx.x < off) {
            s1[threadIdx.x] += s1[threadIdx.x + off];
            s2[threadIdx.x] += s2[threadIdx.x + off];
        }
        __syncthreads();
    }
    if (threadIdx.x == 0) {
        const float fb = (float)(NB > 0 ? NB : 1);
        outp[0] = s1[0] / fb + s2[0] / (fb * (float)(Bn > 0 ? Bn : 1));
    }
}

extern "C" void kernel_launch(void* const* d_in, const int* in_sizes, int n_in,
                              void* d_out, int out_size, void* d_ws, size_t ws_size,
                              hipStream_t stream) {
    const float* outv    = (const float*)d_in[0];   // (B,1,64,64,64)
    const float* q       = (const float*)d_in[1];   // (NQ,)
    const float* xq      = (const float*)d_in[2];   // (NQ,3)
    // d_in[3] = points: not needed (greens only evaluated at boundary indices)
    const int*   xi      = (const int*)d_in[4];
    const int*   yi      = (const int*)d_in[5];
    const int*   zi      = (const int*)d_in[6];
    const float* normals = (const float*)d_in[7];

    const int NB = in_sizes[4];
    const int NQ = in_sizes[1];
    const int Bn = in_sizes[0] / VOL;

    int nblocks = (NB + 15) / 16;
    if (nblocks < 1) nblocks = 1;
    float* partials = (float*)d_ws;

    ibl_greens_kernel<<<nblocks, 32, 0, stream>>>(outv, q, xq, xi, yi, zi,
                                                  normals, partials, NB, NQ, Bn);
    ibl_reduce_kernel<<<1, 256, 0, stream>>>(partials, nblocks, NB, Bn,
                                             (float*)d_out);
}